// chebyshevConv_61701500174788
// MI455X (gfx1250) — compile-verified
//
#include <hip/hip_runtime.h>
#include <math.h>

// Problem constants (from reference)
#define N_    8
#define M_    49152
#define FIN_  64
#define K_    6
#define OUT_  64
#define E_    393216
#define CFN_  (FIN_ * N_)            // 512  columns of each Chebyshev buffer
#define KK_   (FIN_ * K_)            // 384  GEMM reduction length
#define R_    (N_ * M_)              // 393216 GEMM rows
#define XS_STRIDE_ ((size_t)CFN_ * (size_t)M_)   // floats per Chebyshev buffer (25,165,824)

typedef float v2f __attribute__((ext_vector_type(2)));
typedef float v8f __attribute__((ext_vector_type(8)));

// ---------------------------------------------------------------------------
// Kernel 1: build x0 in COLUMN-MAJOR layout: xs0[(f*8+n)*M + m] = x[n,m,f]
// Column-major so the WMMA A-fragment loads (lanes 0..15 = consecutive m) are
// fully coalesced 64B transactions in the GEMM.
// ---------------------------------------------------------------------------
__global__ void build_x0_kernel(const float* __restrict__ x, float* __restrict__ x0)
{
    size_t idx = (size_t)blockIdx.x * blockDim.x + threadIdx.x;   // over CFN_*M_
    if (idx >= XS_STRIDE_) return;
    int m = (int)(idx % M_);
    int c = (int)(idx / M_);          // c = f*8 + n
    int f = c >> 3;
    int n = c & 7;
    x0[idx] = x[((size_t)n * M_ + (size_t)m) * FIN_ + f];
}

// ---------------------------------------------------------------------------
// Kernel 2a: dst = 0   /   dst = -src   (Chebyshev recurrence seed)
// ---------------------------------------------------------------------------
__global__ void init_zero_kernel(float* __restrict__ dst)
{
    size_t idx = (size_t)blockIdx.x * blockDim.x + threadIdx.x;
    if (idx < XS_STRIDE_) dst[idx] = 0.0f;
}

__global__ void init_neg_kernel(const float* __restrict__ src, float* __restrict__ dst)
{
    size_t idx = (size_t)blockIdx.x * blockDim.x + threadIdx.x;
    if (idx < XS_STRIDE_) dst[idx] = -src[idx];
}

// ---------------------------------------------------------------------------
// Kernel 2b: edge-scatter spmm:  dst[c, rows[e]] += alpha*vals[e]*src[c, cols[e]]
// Both src/dst buffers (100 MB each) are L2-resident on MI455X (192 MB L2),
// so fp32 hardware atomics (global_atomic_add_f32) are the cheapest path.
// Each thread handles one edge x 4 feature-columns to amortize index loads.
// ---------------------------------------------------------------------------
__global__ void spmm_scatter_kernel(const float* __restrict__ src,
                                    float* __restrict__ dst,
                                    const int* __restrict__ rows,
                                    const int* __restrict__ cols,
                                    const float* __restrict__ vals,
                                    float alpha)
{
    int e = blockIdx.x * blockDim.x + threadIdx.x;
    if (e >= E_) return;
    int c0 = blockIdx.y * 4;
    float v = alpha * vals[e];
    int r  = rows[e];
    int cc = cols[e];
#pragma unroll
    for (int j = 0; j < 4; ++j) {
        size_t off = (size_t)(c0 + j) * M_;
        unsafeAtomicAdd(&dst[off + (size_t)r], v * src[off + (size_t)cc]);
    }
}

// ---------------------------------------------------------------------------
// Kernel 3: fp32 WMMA GEMM + bias + ELU.
//   out[row = n*M+m, o] = ELU( sum_{f,k} xs[k][(f*8+n)*M+m] * W[f*6+k, o] + b[o] )
//
// One wave owns a 32-row x 64-col strip: 8 x v8f accumulators (2 row-fragments
// x 4 col-tiles). The reduction column index col = kk + 2*hi + j advances by 4
// per WMMA step; (col%6, col/6) has period 3 steps (12 cols = 2 'f' values),
// so the loop is 32 outer iterations of 3 k-steps with SIX per-lane A stream
// pointers, each bumped by a constant 16*M floats per iteration -> no div/mod
// in the loop. The 2nd row-fragment (rows m+16) is the SAME pointers at a
// +64-byte immediate offset (column-major layout), and it reuses the SAME B
// fragments -> per outer body: 12 global b32 + 12 ds b64 + 6 ptr adds for
// 24 v_wmma_f32_16x16x4_f32.
// ---------------------------------------------------------------------------
#define ADV_ ((size_t)16 * M_)        // A stream advance: 12 cols = 2 f's = 16*M floats

__global__ void __launch_bounds__(256)
cheb_gemm_elu_kernel(const float* __restrict__ xs,      // 6 column-major buffers
                     const float* __restrict__ weight,  // [384][64] row-major
                     const float* __restrict__ bias,    // [64]
                     float* __restrict__ out)           // [N, M, OUT] row-major
{
    __shared__ float wlds[OUT_ * KK_];                  // transposed [o][kr], 96 KB

    // Cooperative load of W, transposing into LDS (coalesced global reads).
    for (int i = threadIdx.x; i < KK_ * OUT_; i += 256) {
        int kr = i / OUT_;
        int o  = i % OUT_;
        wlds[o * KK_ + kr] = weight[i];
    }
    __syncthreads();

    const int wave = threadIdx.x >> 5;
    const int lane = threadIdx.x & 31;
    const int ln   = lane & 15;     // A: row-in-tile, B/D: column-in-tile
    const int hi   = lane >> 4;     // K half selector for A/B fragments

    const int strip = blockIdx.x * 8 + wave;     // R_/32 strips total
    const int row0  = strip << 5;                // 32 rows per wave
    const int n     = row0 / M_;                 // uniform across strip (M%32==0)
    const int m     = (row0 % M_) + ln;
    const size_t pm = (size_t)n * M_ + (size_t)m;

    // Six A stream pointers: slot (s,j) covers col = 4*s + 2*hi + j at kk=0.
    const float* pA[6];
#pragma unroll
    for (int s = 0; s < 3; ++s) {
#pragma unroll
        for (int j = 0; j < 2; ++j) {
            int c  = 4 * s + 2 * hi + j;         // 0..11
            int f  = c / K_;
            int kc = c - f * K_;
            pA[2 * s + j] = xs + (size_t)kc * XS_STRIDE_ + (size_t)(f * 8) * M_ + pm;
        }
    }

    v8f accA0 = {}, accA1 = {}, accA2 = {}, accA3 = {};   // rows m .. m+15
    v8f accB0 = {}, accB1 = {}, accB2 = {}, accB3 = {};   // rows m+16 .. m+31

    for (int kk = 0; kk < KK_; kk += 12) {
#pragma unroll
        for (int s = 0; s < 3; ++s) {
            // A fragments: rows (m, m+16) x cols (col0, col0+1)
            v2f a0, a1;
            a0.x = pA[2 * s + 0][0];
            a0.y = pA[2 * s + 1][0];
            a1.x = pA[2 * s + 0][16];            // +64B immediate offset
            a1.y = pA[2 * s + 1][16];

            // B fragments from LDS (shared by both row-fragments)
            int bk = kk + 4 * s + 2 * hi;
            v2f b0 = *(const v2f*)&wlds[(0 * 16 + ln) * KK_ + bk];
            v2f b1 = *(const v2f*)&wlds[(1 * 16 + ln) * KK_ + bk];
            v2f b2 = *(const v2f*)&wlds[(2 * 16 + ln) * KK_ + bk];
            v2f b3 = *(const v2f*)&wlds[(3 * 16 + ln) * KK_ + bk];

            accA0 = __builtin_amdgcn_wmma_f32_16x16x4_f32(false, a0, false, b0, (short)0, accA0, false, false);
            accA1 = __builtin_amdgcn_wmma_f32_16x16x4_f32(false, a0, false, b1, (short)0, accA1, false, false);
            accA2 = __builtin_amdgcn_wmma_f32_16x16x4_f32(false, a0, false, b2, (short)0, accA2, false, false);
            accA3 = __builtin_amdgcn_wmma_f32_16x16x4_f32(false, a0, false, b3, (short)0, accA3, false, false);
            accB0 = __builtin_amdgcn_wmma_f32_16x16x4_f32(false, a1, false, b0, (short)0, accB0, false, false);
            accB1 = __builtin_amdgcn_wmma_f32_16x16x4_f32(false, a1, false, b1, (short)0, accB1, false, false);
            accB2 = __builtin_amdgcn_wmma_f32_16x16x4_f32(false, a1, false, b2, (short)0, accB2, false, false);
            accB3 = __builtin_amdgcn_wmma_f32_16x16x4_f32(false, a1, false, b3, (short)0, accB3, false, false);
        }
        // Bump the six A stream pointers by a constant stride.
#pragma unroll
        for (int t = 0; t < 6; ++t) pA[t] += ADV_;
    }

    // ---- Epilogue: bias + ELU + coalesced b32 stores.
    // Fragment element v of lane l -> (rowbase + v + 8*hi, c*16 + ln)
    float bv0 = bias[0 * 16 + ln];
    float bv1 = bias[1 * 16 + ln];
    float bv2 = bias[2 * 16 + ln];
    float bv3 = bias[3 * 16 + ln];

#pragma unroll
    for (int v = 0; v < 8; ++v) {
        size_t rowA = (size_t)(row0 + v + 8 * hi);
        size_t rowB = rowA + 16;
        float r0 = accA0[v] + bv0, r1 = accA1[v] + bv1;
        float r2 = accA2[v] + bv2, r3 = accA3[v] + bv3;
        float s0 = accB0[v] + bv0, s1 = accB1[v] + bv1;
        float s2 = accB2[v] + bv2, s3 = accB3[v] + bv3;
        r0 = (r0 > 0.0f) ? r0 : (__expf(r0) - 1.0f);
        r1 = (r1 > 0.0f) ? r1 : (__expf(r1) - 1.0f);
        r2 = (r2 > 0.0f) ? r2 : (__expf(r2) - 1.0f);
        r3 = (r3 > 0.0f) ? r3 : (__expf(r3) - 1.0f);
        s0 = (s0 > 0.0f) ? s0 : (__expf(s0) - 1.0f);
        s1 = (s1 > 0.0f) ? s1 : (__expf(s1) - 1.0f);
        s2 = (s2 > 0.0f) ? s2 : (__expf(s2) - 1.0f);
        s3 = (s3 > 0.0f) ? s3 : (__expf(s3) - 1.0f);
        out[rowA * OUT_ + 0 * 16 + ln] = r0;
        out[rowA * OUT_ + 1 * 16 + ln] = r1;
        out[rowA * OUT_ + 2 * 16 + ln] = r2;
        out[rowA * OUT_ + 3 * 16 + ln] = r3;
        out[rowB * OUT_ + 0 * 16 + ln] = s0;
        out[rowB * OUT_ + 1 * 16 + ln] = s1;
        out[rowB * OUT_ + 2 * 16 + ln] = s2;
        out[rowB * OUT_ + 3 * 16 + ln] = s3;
    }
}

// ---------------------------------------------------------------------------
extern "C" void kernel_launch(void* const* d_in, const int* in_sizes, int n_in,
                              void* d_out, int out_size, void* d_ws, size_t ws_size,
                              hipStream_t stream)
{
    (void)in_sizes; (void)n_in; (void)out_size; (void)ws_size;

    const float* x      = (const float*)d_in[0];
    const int*   rows   = (const int*)  d_in[1];
    const int*   cols   = (const int*)  d_in[2];
    const float* vals   = (const float*)d_in[3];
    const float* weight = (const float*)d_in[4];
    const float* bias   = (const float*)d_in[5];
    float*       out    = (float*)d_out;
    float*       xs     = (float*)d_ws;   // needs 6 * XS_STRIDE_ * 4 B = 604 MB

    const int elem_blocks = (int)(XS_STRIDE_ / 256);   // 98304, exact

    // T_0 = x0 (column-major)
    build_x0_kernel<<<elem_blocks, 256, 0, stream>>>(x, xs);

    // Chebyshev recurrence: T_1 = L x0 ; T_k = 2 L T_{k-1} - T_{k-2}
    for (int s = 1; s < K_; ++s) {
        float*       dst = xs + (size_t)s * XS_STRIDE_;
        const float* src = xs + (size_t)(s - 1) * XS_STRIDE_;
        if (s == 1) {
            init_zero_kernel<<<elem_blocks, 256, 0, stream>>>(dst);
        } else {
            const float* prev2 = xs + (size_t)(s - 2) * XS_STRIDE_;
            init_neg_kernel<<<elem_blocks, 256, 0, stream>>>(prev2, dst);
        }
        spmm_scatter_kernel<<<dim3(E_ / 256, CFN_ / 4), 256, 0, stream>>>(
            src, dst, rows, cols, vals, (s == 1) ? 1.0f : 2.0f);
    }

    // fp32 WMMA GEMM + bias + ELU   (R_/32 strips, 8 waves/block)
    cheb_gemm_elu_kernel<<<R_ / 32 / 8, 256, 0, stream>>>(xs, weight, bias, out);
}